// TextDetectionModel_27127013441992
// MI455X (gfx1250) — compile-verified
//
#include <hip/hip_runtime.h>

typedef __attribute__((ext_vector_type(2))) float v2f;
typedef __attribute__((ext_vector_type(8))) float v8f;

#define NPART_MAX 1024

// BCE with the reference's -100 log clamp. __logf -> v_log_f32 (TRANS unit),
// keeping the elementwise pass bandwidth-bound instead of VALU-bound.
__device__ __forceinline__ float bce(float p, float t) {
    float lp  = fmaxf(__logf(p),        -100.0f);
    float l1p = fmaxf(__logf(1.0f - p), -100.0f);
    return -(t * lp + (1.0f - t) * l1p);
}

// 32-lane wave reduction through the CDNA5 tensor core.
// A (16x4 f32): lane L<16 supplies A[L][0]=v (VGPR0), A[L][1]=0 (VGPR1);
// lane L>=16 supplies A[L-16][2]=v, A[L-16][3]=0 (per ISA 16x4 f32 A layout).
// B (4x16) = all ones, so D[m][n] = v[m] + v[m+16] for every column n.
// D layout: lane n<16 holds D[0..7][n] in 8 VGPRs, lane n+16 holds D[8..15][n];
// summing a lane's 8 values and adding the xor-16 partner gives the full
// 32-lane total replicated in every lane. EXEC must be all ones (it is: all
// callers are convergent).
__device__ __forceinline__ float wave_reduce_wmma(float v) {
    v2f a; a.x = v;    a.y = 0.0f;
    v2f b; b.x = 1.0f; b.y = 1.0f;
    v8f c = {0.0f, 0.0f, 0.0f, 0.0f, 0.0f, 0.0f, 0.0f, 0.0f};
    v8f d = __builtin_amdgcn_wmma_f32_16x16x4_f32(false, a, false, b,
                                                  (short)0, c, false, false);
    float s = ((d[0] + d[1]) + (d[2] + d[3])) + ((d[4] + d[5]) + (d[6] + d[7]));
    s += __shfl_xor(s, 16, 32);
    return s;
}

// Deterministic block reduction: WMMA wave reduce, then fixed-order LDS sum.
__device__ __forceinline__ float block_reduce(float v, float* lds) {
    float s = wave_reduce_wmma(v);
    int wave = threadIdx.x >> 5;
    int lane = threadIdx.x & 31;
    if (lane == 0) lds[wave] = s;
    __syncthreads();
    float total = 0.0f;
    int nwaves = (int)(blockDim.x >> 5);
    for (int w = 0; w < nwaves; ++w) total += lds[w];   // same order every call
    __syncthreads();                                     // allow lds reuse
    return total;
}

// ---- Kernel 1: BCE over the 16x1x512x512 maps, one partial per block. ----
__global__ __launch_bounds__(256) void k_text_map_partial(
    const float4* __restrict__ pred, const float4* __restrict__ tgt,
    float* __restrict__ partial, int n4) {
    __shared__ float lds[8];
    float acc = 0.0f;
    int idx    = blockIdx.x * blockDim.x + threadIdx.x;
    int stride = gridDim.x * blockDim.x;
    for (int i = idx; i < n4; i += stride) {
        float4 p = pred[i];
        float4 t = tgt[i];
        acc += (bce(p.x, t.x) + bce(p.y, t.y)) + (bce(p.z, t.z) + bce(p.w, t.w));
    }
    float s = block_reduce(acc, lds);
    if (threadIdx.x == 0) partial[blockIdx.x] = s;
}

// ---- Kernel 2: reduce partials + full box branch + final 4 outputs. ----
__global__ __launch_bounds__(256) void k_finalize(
    const float* __restrict__ partial, int npart,
    const float* __restrict__ conf_pred,
    const float* __restrict__ bbox_pred,
    const float* __restrict__ box_conf,
    const int4*  __restrict__ box_xyxy,
    const unsigned char* __restrict__ box_mask,
    float* __restrict__ out,
    int B, int H, int W, int N, float inv_count) {
    __shared__ float lds[8];

    float tacc = 0.0f;
    for (int i = threadIdx.x; i < npart; i += blockDim.x) tacc += partial[i];

    float blacc = 0.0f, clacc = 0.0f, vacc = 0.0f;
    const int total = B * N;
    const float fW = (float)W, fH = (float)H;
    const size_t plane = (size_t)H * (size_t)W;
    for (int i = threadIdx.x; i < total; i += blockDim.x) {
        int b = i / N;
        int4 q = box_xyxy[i];                     // x1,y1,x2,y2
        int cx = (q.x + q.z) >> 1;                // floor((x1+x2)/2)
        int cy = (q.y + q.w) >> 1;
        bool valid = (box_mask[i] != 0) & (cx >= 0) & (cx < W) & (cy >= 0) & (cy < H);
        int cxc = min(max(cx, 0), W - 1);
        int cyc = min(max(cy, 0), H - 1);

        float pc = conf_pred[(size_t)b * plane + (size_t)cyc * W + cxc];
        float tb[4] = { q.x / fW, q.y / fH, q.z / fW, q.w / fH };
        size_t base = (size_t)b * 4 * plane + (size_t)cyc * W + cxc;
        float l1 = 0.0f;
        for (int ch = 0; ch < 4; ++ch) {
            float pb = bbox_pred[base + (size_t)ch * plane];
            float d  = fabsf(pb - tb[ch]);
            l1 += (d < 1.0f) ? 0.5f * d * d : (d - 0.5f);
        }
        l1 *= 0.25f;                              // mean over 4 channels
        float cl = bce(pc, box_conf[i]);
        float vf = valid ? 1.0f : 0.0f;
        blacc += l1 * vf;
        clacc += cl * vf;
        vacc  += vf;
    }

    float tsum = block_reduce(tacc,  lds);
    float bsum = block_reduce(blacc, lds);
    float csum = block_reduce(clacc, lds);
    float vsum = block_reduce(vacc,  lds);

    if (threadIdx.x == 0) {
        float text_loss = tsum * inv_count;
        float denom = fmaxf(vsum, 1.0f);
        bool  any   = vsum > 0.0f;
        float box_loss  = any ? (bsum / denom) : 0.0f;
        float conf_loss = any ? (csum / denom) : 0.0f;
        out[0] = text_loss + box_loss + conf_loss;
        out[1] = text_loss;
        out[2] = box_loss;
        out[3] = conf_loss;
    }
}

extern "C" void kernel_launch(void* const* d_in, const int* in_sizes, int n_in,
                              void* d_out, int out_size, void* d_ws, size_t ws_size,
                              hipStream_t stream) {
    const float* text_pred = (const float*)d_in[0];   // (16,1,512,512) f32
    const float* text_tgt  = (const float*)d_in[1];   // (16,1,512,512) f32
    const float* conf_pred = (const float*)d_in[2];   // (16,1,512,512) f32
    const float* bbox_pred = (const float*)d_in[3];   // (16,4,512,512) f32
    const float* box_conf  = (const float*)d_in[4];   // (16,256)       f32
    const int*   box_xyxy  = (const int*)d_in[5];     // (16,256,4)     i32
    const unsigned char* box_mask = (const unsigned char*)d_in[6]; // (16,256) bool

    const int B = 16, H = 512, W = 512, N = 256;
    const int n_map = in_sizes[0];                    // B*H*W = 4194304
    const int n4 = n_map / 4;

    int npart = (int)(ws_size / sizeof(float));
    if (npart > NPART_MAX) npart = NPART_MAX;
    if (npart < 1) npart = 1;
    float* partial = (float*)d_ws;

    k_text_map_partial<<<npart, 256, 0, stream>>>(
        (const float4*)text_pred, (const float4*)text_tgt, partial, n4);

    k_finalize<<<1, 256, 0, stream>>>(
        partial, npart, conf_pred, bbox_pred, box_conf,
        (const int4*)box_xyxy, box_mask, (float*)d_out,
        B, H, W, N, 1.0f / (float)n_map);
}